// Encoder_9328668967786
// MI455X (gfx1250) — compile-verified
//
#include <hip/hip_runtime.h>

#define N_NODES 100000
#define E_EDGES 1600000
#define NFEAT   256
#define NHID    128
#define LATENT  64

typedef float v2f __attribute__((ext_vector_type(2)));
typedef float v8f __attribute__((ext_vector_type(8)));

// ---------------------------------------------------------------------------
// Pack W[K x NT*16] (row-major) into WMMA-B fragment order:
//   Wpk[((s*NT + t)*32 + lane)] = { W[(4s+ksel)*LD + t*16 + col],
//                                   W[(4s+ksel+1)*LD + t*16 + col] }
// with ksel = (lane>>4)*2, col = lane&15. One float2 per (k-step, tile, lane)
// so the GEMM reads each B fragment as a single coalesced b64.
// ---------------------------------------------------------------------------
__global__ __launch_bounds__(256)
void pack_w(const float* __restrict__ W, v2f* __restrict__ Wpk, int K, int NT) {
    int i = blockIdx.x * blockDim.x + threadIdx.x;
    int total = (K / 4) * NT * 32;
    if (i >= total) return;
    int lane = i & 31;
    int t    = (i >> 5) % NT;
    int s    = (i >> 5) / NT;
    int ksel = (lane >> 4) << 1;
    int col  = lane & 15;
    int LD   = NT * 16;
    int k    = 4 * s + ksel;
    v2f v;
    v.x = W[(size_t)k * LD + t * 16 + col];
    v.y = W[(size_t)(k + 1) * LD + t * 16 + col];
    Wpk[i] = v;
}

// ---------------------------------------------------------------------------
// C[M x NT*16] = A[M x K] @ W (+bias). W pre-packed in fragment order.
// One wave computes MT adjacent 16-row strips, reusing each B fragment MT
// times. WMMA f32 16x16x4; M % (16*MT*2-per-block granularity) handled by a
// wave-uniform guard, so EXEC is all-ones at every WMMA.
// ---------------------------------------------------------------------------
template<int NT, int MT>
__global__ __launch_bounds__(64)
void gemm_wmma_f32(const float* __restrict__ A, const v2f* __restrict__ Wpk,
                   const float* __restrict__ bias, float* __restrict__ C,
                   int M, int K) {
    const int lane  = threadIdx.x & 31;
    const int wave  = threadIdx.x >> 5;
    const int tile0 = (blockIdx.x * 2 + wave) * MT;
    if (tile0 * 16 >= M) return;                 // wave-uniform
    const int kSel = (lane >> 4) << 1;           // lanes 0-15 -> K 0,1 ; 16-31 -> K 2,3
    const int col  = lane & 15;
    constexpr int LD = NT * 16;

    const float* Ap[MT];
#pragma unroll
    for (int m = 0; m < MT; ++m)
        Ap[m] = A + (size_t)(tile0 * 16 + m * 16 + col) * K + kSel;

    const v2f* wp = Wpk + lane;

    v8f acc[MT][NT] = {};

    for (int k0 = 0; k0 < K; k0 += 4) {
        v2f a[MT];
#pragma unroll
        for (int m = 0; m < MT; ++m)
            a[m] = *(const v2f*)(Ap[m] + k0);    // single b64, 8B-aligned
        const v2f* wk = wp + (size_t)(k0 >> 2) * NT * 32;
#pragma unroll
        for (int t = 0; t < NT; ++t) {
            v2f b = wk[t * 32];                  // single coalesced b64
#pragma unroll
            for (int m = 0; m < MT; ++m)
                acc[m][t] = __builtin_amdgcn_wmma_f32_16x16x4_f32(
                    false, a[m], false, b, (short)0, acc[m][t], false, false);
        }
    }

    const int rsel = (lane >> 4) << 3;           // lanes 16-31 hold rows M=8..15
#pragma unroll
    for (int m = 0; m < MT; ++m) {
        const int row0 = (tile0 + m) * 16;
#pragma unroll
        for (int t = 0; t < NT; ++t) {
            const float bv = bias ? bias[t * 16 + col] : 0.0f;
#pragma unroll
            for (int v = 0; v < 8; ++v) {
                C[(size_t)(row0 + rsel + v) * LD + t * 16 + col] = acc[m][t][v] + bv;
            }
        }
    }
}

// out[row[e]] += w[e] * S[col[e]]   (32 threads/edge, float4 per thread)
__global__ __launch_bounds__(256)
void spmm_scatter(const int* __restrict__ row, const int* __restrict__ col,
                  const float* __restrict__ w, const float* __restrict__ S,
                  float* __restrict__ out) {
    long long tid = (long long)blockIdx.x * blockDim.x + threadIdx.x;
    const int e = (int)(tid >> 5);               // NHID/4 = 32 chunks per edge
    if (e >= E_EDGES) return;
    const int f = ((int)tid & 31) << 2;
    const int r = row[e];
    const int c = col[e];
    const float wt = w[e];
    const float4 v = *reinterpret_cast<const float4*>(S + (size_t)c * NHID + f);
    float* o = out + (size_t)r * NHID + f;
    unsafeAtomicAdd(o + 0, v.x * wt);
    unsafeAtomicAdd(o + 1, v.y * wt);
    unsafeAtomicAdd(o + 2, v.z * wt);
    unsafeAtomicAdd(o + 3, v.w * wt);
}

__global__ __launch_bounds__(256)
void bias_relu(float* __restrict__ h, const float* __restrict__ b, int total) {
    int i = blockIdx.x * blockDim.x + threadIdx.x;
    if (i >= total) return;
    float v = h[i] + b[i & (NHID - 1)];
    h[i] = v > 0.0f ? v : 0.0f;
}

extern "C" void kernel_launch(void* const* d_in, const int* in_sizes, int n_in,
                              void* d_out, int out_size, void* d_ws, size_t ws_size,
                              hipStream_t stream) {
    const float* x   = (const float*)d_in[0];
    const int*   ei  = (const int*)  d_in[1];   // [2, E] flat
    const float* ew  = (const float*)d_in[2];
    const float* g1w = (const float*)d_in[3];
    const float* g1b = (const float*)d_in[4];
    const float* g2w = (const float*)d_in[5];
    const float* g2b = (const float*)d_in[6];
    const float* muw = (const float*)d_in[7];
    const float* mub = (const float*)d_in[8];
    const float* lvw = (const float*)d_in[9];
    const float* lvb = (const float*)d_in[10];
    float* out = (float*)d_out;

    const int* rowp = ei;
    const int* colp = ei + E_EDGES;

    // workspace layout
    float* bufA = (float*)d_ws;                       // support   (N x 128), 51.2 MB
    float* bufB = bufA + (size_t)N_NODES * NHID;      // aggregate (N x 128), 51.2 MB
    v2f* pk1 = (v2f*)(bufB + (size_t)N_NODES * NHID); // packed gc1_w : 16384 v2f
    v2f* pk2 = pk1 + (NFEAT / 4) * (NHID / 16) * 32;  // packed gc2_w :  8192 v2f
    v2f* pkM = pk2 + (NHID  / 4) * (NHID / 16) * 32;  // packed mu_w  :  4096 v2f
    v2f* pkL = pkM + (NHID  / 4) * (LATENT / 16) * 32;// packed lv_w  :  4096 v2f

    const size_t hbytes = (size_t)N_NODES * NHID * sizeof(float);

    // ---- pack weights into WMMA-B fragment order (tiny, L2-resident)
    pack_w<<<(NFEAT / 4) * (NHID / 16) * 32 / 256, 256, 0, stream>>>(g1w, pk1, NFEAT, NHID / 16);
    pack_w<<<(NHID  / 4) * (NHID / 16) * 32 / 256, 256, 0, stream>>>(g2w, pk2, NHID, NHID / 16);
    pack_w<<<(NHID  / 4) * (LATENT / 16) * 32 / 256, 256, 0, stream>>>(muw, pkM, NHID, LATENT / 16);
    pack_w<<<(NHID  / 4) * (LATENT / 16) * 32 / 256, 256, 0, stream>>>(lvw, pkL, NHID, LATENT / 16);

    // grids: 6250 M-tiles, 2 tiles/wave, 2 waves/block -> ceil(6250/4) = 1563
    const int mtiles = N_NODES / 16;                  // 6250
    dim3 ggrid((mtiles + 3) / 4);
    dim3 gblk(64);

    const long long etot = (long long)E_EDGES * (NHID / 4);
    const int eblocks = (int)((etot + 255) / 256);
    const int hblocks = (N_NODES * NHID) / 256;       // 50000

    // ---- layer 1: support = x @ gc1_w ; agg = A @ support ; h = relu(agg + b1)
    gemm_wmma_f32<NHID/16, 2><<<ggrid, gblk, 0, stream>>>(x, pk1, nullptr, bufA, N_NODES, NFEAT);
    hipMemsetAsync(bufB, 0, hbytes, stream);
    spmm_scatter<<<eblocks, 256, 0, stream>>>(rowp, colp, ew, bufA, bufB);
    bias_relu<<<hblocks, 256, 0, stream>>>(bufB, g1b, N_NODES * NHID);

    // ---- layer 2: support = h @ gc2_w ; agg = A @ support ; h = relu(agg + b2)
    gemm_wmma_f32<NHID/16, 2><<<ggrid, gblk, 0, stream>>>(bufB, pk2, nullptr, bufA, N_NODES, NHID);
    hipMemsetAsync(bufB, 0, hbytes, stream);
    spmm_scatter<<<eblocks, 256, 0, stream>>>(rowp, colp, ew, bufA, bufB);
    bias_relu<<<hblocks, 256, 0, stream>>>(bufB, g2b, N_NODES * NHID);

    // ---- heads: mu = h @ mu_w + mu_b ; log_var = h @ lv_w + lv_b
    gemm_wmma_f32<LATENT/16, 2><<<ggrid, gblk, 0, stream>>>(bufB, pkM, mub, out, N_NODES, NHID);
    gemm_wmma_f32<LATENT/16, 2><<<ggrid, gblk, 0, stream>>>(bufB, pkL, lvb,
                                                            out + (size_t)N_NODES * LATENT,
                                                            N_NODES, NHID);
}